// PostProcess_137438953520
// MI455X (gfx1250) — compile-verified
//
#include <hip/hip_runtime.h>
#include <stdint.h>

#define N_IMG 32
#define N_CLS 81
#define CM1   80
#define N_BOX 8732
#define TOPK  400
#define MAXOUT 100
#define TILE_B 192
#define CAP_MAX (N_BOX * 19)

#define F_SCALE_XY 0.1f
#define F_SCALE_WH 0.2f
#define F_THRESH   0.05f
#define F_CRIT     0.5f
#define F_MINSZ    (1.0f / 300.0f)

#define USE_ASYNC_LDS 1

#if defined(__HIP_DEVICE_COMPILE__) && __has_builtin(__builtin_amdgcn_wmma_f32_16x16x4_f32)
#define HAVE_WMMA_F32 1
#endif

typedef __attribute__((ext_vector_type(2))) float v2f;
typedef __attribute__((ext_vector_type(8))) float v8f;

// ---------------------------------------------------------------- helpers
__device__ __forceinline__ void wait_async0() {
#if defined(__HIP_DEVICE_COMPILE__)
#if __has_builtin(__builtin_amdgcn_s_wait_asynccnt)
  __builtin_amdgcn_s_wait_asynccnt(0);
#else
  asm volatile("s_wait_asynccnt 0" ::: "memory");
#endif
#endif
}

__device__ __forceinline__ void async_load_b128(const float* g, float* lds_generic) {
#if defined(__HIP_DEVICE_COMPILE__) && defined(__gfx1250__) && USE_ASYNC_LDS
  uint32_t lofs = (uint32_t)(uintptr_t)lds_generic;  // low 32 bits = LDS offset
  unsigned long long ga = (unsigned long long)(uintptr_t)g;
  asm volatile("global_load_async_to_lds_b128 %0, %1, off" ::"v"(lofs), "v"(ga) : "memory");
#else
  (void)g; (void)lds_generic;
#endif
}

// ---------------------------------------------------------------- kernel 1: box decode
__global__ void k_decode(const float* __restrict__ bb, const float* __restrict__ db,
                         float4* __restrict__ boxes4) {
  int i = blockIdx.x * blockDim.x + threadIdx.x;
  if (i >= N_IMG * N_BOX) return;
  int n = i / N_BOX, b = i - n * N_BOX;
  const float* base = bb + (size_t)n * 4 * N_BOX + b;
  float bx = base[0];
  float by = base[(size_t)N_BOX];
  float bw = base[(size_t)2 * N_BOX];
  float bh = base[(size_t)3 * N_BOX];
  float dx = db[b * 4 + 0], dy = db[b * 4 + 1], dw = db[b * 4 + 2], dh = db[b * 4 + 3];
  float x = F_SCALE_XY * bx * dw + dx;
  float y = F_SCALE_XY * by * dh + dy;
  float w = expf(F_SCALE_WH * bw) * dw;
  float h = expf(F_SCALE_WH * bh) * dh;
  float l = fminf(fmaxf(x - 0.5f * w, 0.f), 1.f);
  float t = fminf(fmaxf(y - 0.5f * h, 0.f), 1.f);
  float r = fminf(fmaxf(x + 0.5f * w, 0.f), 1.f);
  float bo = fminf(fmaxf(y + 0.5f * h, 0.f), 1.f);
  boxes4[i] = make_float4(l, t, r, bo);
}

// ---------------------------------------------------------------- kernel 2: softmax + gather
// block = 192 threads (6 full wave32), one (image, 192-box tile) per block.
__global__ void k_scores(const float* __restrict__ sc, const float4* __restrict__ boxes4,
                         float* __restrict__ candVal, int* __restrict__ candIdx,
                         int* __restrict__ candCnt, int cap) {
  __shared__ float s[N_CLS * TILE_B];  // 62,208 bytes
  __shared__ float maxv[TILE_B];
  __shared__ float sums[TILE_B];
  const int n = blockIdx.y;
  const int b0 = blockIdx.x * TILE_B;
  const int tileW = min(TILE_B, N_BOX - b0);  // 192 or 92, both %4==0
  const float* base = sc + (size_t)n * N_CLS * N_BOX + b0;
  const int tid = threadIdx.x;

  const int chunksPerRow = tileW >> 2;  // 16B chunks per class row
  const int totalChunks = N_CLS * chunksPerRow;
#if USE_ASYNC_LDS && defined(__gfx1250__)
  for (int q = tid; q < totalChunks; q += blockDim.x) {
    int c = q / chunksPerRow, k = q - c * chunksPerRow;
    async_load_b128(base + (size_t)c * N_BOX + (k << 2), &s[c * TILE_B + (k << 2)]);
  }
  wait_async0();
#else
  for (int q = tid; q < totalChunks; q += blockDim.x) {
    int c = q / chunksPerRow, k = q - c * chunksPerRow;
    const float4 v = *(const float4*)(base + (size_t)c * N_BOX + (k << 2));
    *(float4*)&s[c * TILE_B + (k << 2)] = v;
  }
#endif
  __syncthreads();

  // ---- per-box max (thread = box) ----
  float m = 0.f;
  if (tid < tileW) {
    m = -3.4e38f;
    for (int c = 0; c < N_CLS; c++) m = fmaxf(m, s[c * TILE_B + tid]);
  }
  maxv[tid] = m;
  __syncthreads();

  // ---- softmax denominator: row-sum of exp via V_WMMA_F32_16X16X4_F32 with B = ones.
  // A (16x4 f32): lanes 0-15 rows M, VGPR0=K0,VGPR1=K1; lanes 16-31 same rows, K2,K3.
  // EXEC must be all ones -> no tid<tileW guard here; padded boxes feed zeros.
  float inv = 0.f;
#ifdef HAVE_WMMA_F32
  {
    const int lane = tid & 31;
    const int wv = tid >> 5;
    const int half = lane >> 4;   // 0 -> K0/K1, 1 -> K2/K3
    const int lrow = lane & 15;   // row within 16-box group
    v2f bones;
    bones.x = 1.f;
    bones.y = 1.f;
    for (int gsel = 0; gsel < 2; gsel++) {
      const int g = wv * 32 + gsel * 16;  // group base box-col within tile
      const int bc = g + lrow;
      const bool live = bc < tileW;
      const float mb = live ? maxv[bc] : 0.f;
      v8f acc = {};
#pragma unroll
      for (int ch = 0; ch < 21; ch++) {  // ceil(81/4) K-chunks
        const int c0 = ch * 4 + half * 2;
        v2f a;
        a.x = (live && c0 < N_CLS) ? expf(s[c0 * TILE_B + bc] - mb) : 0.f;
        a.y = (live && (c0 + 1) < N_CLS) ? expf(s[(c0 + 1) * TILE_B + bc] - mb) : 0.f;
        acc = __builtin_amdgcn_wmma_f32_16x16x4_f32(false, a, false, bones, (short)0, acc,
                                                    false, false);
      }
      // D col 0: lane 0 VGPR r = sum(box g+r); lane 16 VGPR r = sum(box g+8+r)
      if (lrow == 0) {
        const int basep = g + half * 8;
#pragma unroll
        for (int r = 0; r < 8; r++) sums[basep + r] = acc[r];
      }
    }
  }
  __syncthreads();
  if (tid < tileW) inv = 1.f / sums[tid];
#else
  if (tid < tileW) {
    float ssum = 0.f;
    for (int c = 0; c < N_CLS; c++) ssum += expf(s[c * TILE_B + tid] - m);
    inv = 1.f / ssum;
    sums[tid] = ssum;
  }
  __syncthreads();
#endif

  // ---- candidate emission (thread = box) ----
  if (tid < tileW) {
    const int b = b0 + tid;
    float4 bx = boxes4[(size_t)n * N_BOX + b];
    const bool vb = (bx.z - bx.x >= F_MINSZ) && (bx.w - bx.y >= F_MINSZ);
    if (vb) {
      for (int c = 1; c < N_CLS; c++) {
        float p = expf(s[c * TILE_B + tid] - m) * inv;
        if (p > F_THRESH) {
          int pos = atomicAdd(&candCnt[n], 1);
          if (pos < cap) {
            candVal[(size_t)n * cap + pos] = p;
            candIdx[(size_t)n * cap + pos] = b * CM1 + (c - 1);
          }
        }
      }
    }
  }
}

// ---------------------------------------------------------------- kernel 3: top-400 per image
__global__ void k_select(float* __restrict__ candVal, const int* __restrict__ candIdx,
                         const int* __restrict__ candCnt, int cap,
                         float* __restrict__ selVal, int* __restrict__ selIdx) {
  const int n = blockIdx.x;
  const int K = min(candCnt[n], cap);
  float* cv = candVal + (size_t)n * cap;
  const int* ci = candIdx + (size_t)n * cap;
  __shared__ float rv[256];
  __shared__ int ri[256];
  __shared__ int rp[256];
  const int tid = threadIdx.x;
  for (int r = 0; r < TOPK; r++) {
    float bestV = -1e30f;
    int bestI = 0x7fffffff, bestP = -1;
    for (int j = tid; j < K; j += blockDim.x) {
      __builtin_prefetch(&cv[j + blockDim.x], 0, 1);
      float v = cv[j];
      int id = ci[j];
      if (v > bestV || (v == bestV && id < bestI)) { bestV = v; bestI = id; bestP = j; }
    }
    rv[tid] = bestV; ri[tid] = bestI; rp[tid] = bestP;
    __syncthreads();
    for (int st = 128; st > 0; st >>= 1) {
      if (tid < st) {
        float v2 = rv[tid + st];
        int i2 = ri[tid + st];
        if (v2 > rv[tid] || (v2 == rv[tid] && i2 < ri[tid])) {
          rv[tid] = v2; ri[tid] = i2; rp[tid] = rp[tid + st];
        }
      }
      __syncthreads();
    }
    if (tid == 0) {
      if (rv[0] > 0.f) {
        selVal[n * TOPK + r] = rv[0];
        selIdx[n * TOPK + r] = ri[0];
        cv[rp[0]] = -1e30f;  // consume
      } else {
        selVal[n * TOPK + r] = -1.f;
        selIdx[n * TOPK + r] = 0;
      }
    }
    __syncthreads();
  }
}

// ---------------------------------------------------------------- kernel 4: NMS + outputs
__global__ void k_nms(const float* __restrict__ selVal, const int* __restrict__ selIdx,
                      const float4* __restrict__ boxes4, float* __restrict__ out) {
  const int n = blockIdx.x;
  __shared__ float cb[TOPK][4];
  __shared__ float ob[TOPK][4];
  __shared__ float area[TOPK];
  __shared__ float sval[TOPK];
  __shared__ int slab[TOPK];
  __shared__ unsigned char keep[TOPK];
  __shared__ int rank[TOPK];

  for (int k = threadIdx.x; k < TOPK; k += blockDim.x) {
    float v = selVal[n * TOPK + k];
    int idx = selIdx[n * TOPK + k];
    int bi = idx / CM1;
    int lab = idx - bi * CM1 + 1;
    float4 bx = boxes4[(size_t)n * N_BOX + bi];
    cb[k][0] = bx.x; cb[k][1] = bx.y; cb[k][2] = bx.z; cb[k][3] = bx.w;
    float off = 2.0f * (float)lab;
    ob[k][0] = bx.x + off; ob[k][1] = bx.y + off;
    ob[k][2] = bx.z + off; ob[k][3] = bx.w + off;
    area[k] = fmaxf(ob[k][2] - ob[k][0], 0.f) * fmaxf(ob[k][3] - ob[k][1], 0.f);
    sval[k] = v;
    slab[k] = lab;
    keep[k] = (v > 0.f) ? 1 : 0;
  }
  __syncthreads();

  for (int i = 0; i < TOPK; i++) {
    if (keep[i]) {  // uniform read
      float lx = ob[i][0], ly = ob[i][1], rx = ob[i][2], ry = ob[i][3], ai = area[i];
      for (int j = i + 1 + threadIdx.x; j < TOPK; j += blockDim.x) {
        if (!keep[j]) continue;
        float ltx = fmaxf(lx, ob[j][0]), lty = fmaxf(ly, ob[j][1]);
        float rbx = fminf(rx, ob[j][2]), rby = fminf(ry, ob[j][3]);
        float iw = fmaxf(rbx - ltx, 0.f), ih = fmaxf(rby - lty, 0.f);
        float inter = iw * ih;
        float iou = inter / fmaxf(ai + area[j] - inter, 1e-12f);
        if (iou > F_CRIT) keep[j] = 0;
      }
    }
    __syncthreads();
  }

  if (threadIdx.x == 0) {
    int c = 0;
    for (int k = 0; k < TOPK; k++) { rank[k] = c; c += keep[k] ? 1 : 0; }
  }
  __syncthreads();

  float* obox = out + (size_t)n * MAXOUT * 4;
  float* olab = out + (size_t)N_IMG * MAXOUT * 4 + (size_t)n * MAXOUT;
  float* osc  = out + (size_t)N_IMG * MAXOUT * 4 + (size_t)N_IMG * MAXOUT + (size_t)n * MAXOUT;
  for (int t = threadIdx.x; t < MAXOUT * 4; t += blockDim.x) obox[t] = 0.f;
  for (int t = threadIdx.x; t < MAXOUT; t += blockDim.x) { olab[t] = 0.f; osc[t] = 0.f; }
  __syncthreads();

  for (int k = threadIdx.x; k < TOPK; k += blockDim.x) {
    if (keep[k]) {
      int r = rank[k];
      if (r < MAXOUT) {
        obox[r * 4 + 0] = cb[k][0];
        obox[r * 4 + 1] = cb[k][1];
        obox[r * 4 + 2] = cb[k][2];
        obox[r * 4 + 3] = cb[k][3];
        olab[r] = (float)slab[k];
        osc[r] = sval[k];
      }
    }
  }
}

// ---------------------------------------------------------------- host
extern "C" void kernel_launch(void* const* d_in, const int* in_sizes, int n_in,
                              void* d_out, int out_size, void* d_ws, size_t ws_size,
                              hipStream_t stream) {
  const float* bboxes = (const float*)d_in[0];  // (32, 4, 8732)
  const float* scores = (const float*)d_in[1];  // (32, 81, 8732)
  const float* dboxes = (const float*)d_in[2];  // (1, 8732, 4)
  float* out = (float*)d_out;                   // 12800 boxes | 3200 labels | 3200 scores

  char* ws = (char*)d_ws;
  size_t o = 0;
  float4* boxes4 = (float4*)(ws + o); o += (size_t)N_IMG * N_BOX * sizeof(float4);
  float* selVal  = (float*)(ws + o);  o += (size_t)N_IMG * TOPK * sizeof(float);
  int*   selIdx  = (int*)(ws + o);    o += (size_t)N_IMG * TOPK * sizeof(int);
  int*   candCnt = (int*)(ws + o);    o += 256;
  size_t remain = (ws_size > o) ? (ws_size - o) : 0;
  long long capll = (long long)(remain / ((size_t)N_IMG * 8));
  int cap = (capll > CAP_MAX) ? CAP_MAX : (int)capll;
  if (cap < 1) cap = 1;
  float* candVal = (float*)(ws + o);  o += (size_t)N_IMG * cap * sizeof(float);
  int*   candIdx = (int*)(ws + o);

  hipMemsetAsync(candCnt, 0, N_IMG * sizeof(int), stream);

  int tot = N_IMG * N_BOX;
  k_decode<<<(tot + 255) / 256, 256, 0, stream>>>(bboxes, dboxes, boxes4);

  dim3 g2((N_BOX + TILE_B - 1) / TILE_B, N_IMG);
  k_scores<<<g2, TILE_B, 0, stream>>>(scores, boxes4, candVal, candIdx, candCnt, cap);

  k_select<<<N_IMG, 256, 0, stream>>>(candVal, candIdx, candCnt, cap, selVal, selIdx);

  k_nms<<<N_IMG, 512, 0, stream>>>(selVal, selIdx, boxes4, out);
}